// ConditionalRandomField_9414568313515
// MI455X (gfx1250) — compile-verified
//
#include <hip/hip_runtime.h>
#include <hip/hip_bf16.h>
#include <math.h>

#define TAGS      128
#define BATCH     256
#define SEQLEN    512
#define START_TAG 126
#define STOP_TAG  127
#define NEGV      (-10000.0f)
#define NB        16      // batch columns per workgroup
#define COLSTR    132     // padded per-column LDS stride (dwords)
#define ESTR      20      // per-wave emission tile column stride (dwords, %4==0)

typedef float v2f __attribute__((ext_vector_type(2)));
typedef float v8f __attribute__((ext_vector_type(8)));

// ---- CDNA5 async global->LDS path (guarded; fallback = direct loads) ------
#if defined(__has_builtin)
#if __has_builtin(__builtin_amdgcn_global_load_async_to_lds_b128)
#define HAVE_ASYNC_LDS 1
#endif
#endif
#ifndef HAVE_ASYNC_LDS
#define HAVE_ASYNC_LDS 0
#endif

#if HAVE_ASYNC_LDS
#if __has_builtin(__builtin_amdgcn_s_wait_asynccnt)
#define WAIT_ASYNC(n) __builtin_amdgcn_s_wait_asynccnt(n)
#else
#define WAIT_ASYNC(n) asm volatile("s_wait_asynccnt " #n ::: "memory")
#endif
// Builtin signature (probed via clang diagnostic): both pointers are to a
// 16-byte int vector; src in global (__device__/AS1), dst in LDS (AS3).
typedef int v4i_t __attribute__((vector_size(16)));
typedef __attribute__((address_space(1))) v4i_t gv4i;
typedef __attribute__((address_space(3))) v4i_t lv4i;
#define ASYNC_B128(gp, lp) \
  __builtin_amdgcn_global_load_async_to_lds_b128((gv4i*)(gp), (lv4i*)(lp), 0, 0)
#endif

__device__ __forceinline__ float hmax16(float v) {
#pragma unroll
  for (int m = 1; m <= 8; m <<= 1) v = fmaxf(v, __shfl_xor(v, m, 32));
  return v;
}
__device__ __forceinline__ float hsum16(float v) {
#pragma unroll
  for (int m = 1; m <= 8; m <<= 1) v += __shfl_xor(v, m, 32);
  return v;
}

// ---------------------------------------------------------------------------
// Forward scan: log-domain recurrence as exp-domain GEMM on f32 WMMA.
// grid = BATCH/NB workgroups x 256 threads (8 wave32 waves).
// Wave w owns output rows [16w,16w+16); its 16x128 exp(trans) tile stays in
// 64 VGPRs for the whole 512-step scan. All 32 B fragments are preloaded to
// registers so the 32 WMMAs (4 independent accumulator chains) issue
// back-to-back behind a single s_wait_dscnt. Emissions are double-buffered
// into LDS with global_load_async_to_lds_b128 (each wave consumes only its
// own async data => no extra barrier, just s_wait_asynccnt).
// ---------------------------------------------------------------------------
__global__ __launch_bounds__(256)
void crf_forward_scan(const float* __restrict__ inputs,
                      const int*   __restrict__ mask,
                      const float* __restrict__ trans,
                      float*       __restrict__ log_den) {
  __shared__ float    fv[NB * COLSTR];            // forward vars [n][j]
  __shared__ float    vb[NB * COLSTR];            // exp(fv - M)  [n][j]
  __shared__ float    Mcol[NB];
  __shared__ float    Scol[NB];
  __shared__ unsigned mbits[NB * (SEQLEN / 32)];  // packed mask bits
#if HAVE_ASYNC_LDS
  __shared__ float    em[2 * 8 * NB * ESTR];      // [buf][wave][col][row]
#endif

  const int tid  = threadIdx.x;
  const int wave = tid >> 5;
  const int lane = tid & 31;
  const int ln   = lane & 15;   // N (batch column) inside WMMA tile
  const int lh   = lane >> 4;   // half-wave select
  const int b0   = blockIdx.x * NB;
  const int irow = 16 * wave + 8 * lh;   // first M row this lane owns

  // Stationary A tiles (V_WMMA_F32_16X16X4_F32 A layout:
  // lanes 0-15: M=lane, vgpr0=K0, vgpr1=K1 ; lanes 16-31: K2,K3).
  v2f A[32];
  {
    const int i = 16 * wave + ln;
#pragma unroll
    for (int kt = 0; kt < 32; ++kt) {
      const int j = 4 * kt + 2 * lh;
      A[kt].x = __expf(trans[i * TAGS + j]);
      A[kt].y = __expf(trans[i * TAGS + j + 1]);
    }
  }

  // pack mask into LDS bitfield: one 32-step word per thread
  {
    const int n = tid >> 4;
    const int w = tid & 15;
    unsigned bw = 0u;
    const int base = (b0 + n) * SEQLEN + w * 32;
#pragma unroll
    for (int b = 0; b < 32; ++b) bw |= (mask[base + b] != 0 ? 1u : 0u) << b;
    mbits[n * (SEQLEN / 32) + w] = bw;
  }

  // init forward variables
  for (int idx = tid; idx < NB * TAGS; idx += 256) {
    const int n = idx >> 7;
    const int j = idx & 127;
    fv[n * COLSTR + j] = (j == START_TAG) ? 0.0f : NEGV;
  }

#if HAVE_ASYNC_LDS
  // prime emissions for t=0 into buffer 0 (own-wave tile only)
  {
    const float* gp = inputs + ((size_t)(b0 + ln) * SEQLEN + 0) * TAGS + irow;
    float* lp = &em[(((size_t)0 * 8 + wave) * NB + ln) * ESTR + 8 * lh];
    ASYNC_B128(gp,     lp);
    ASYNC_B128(gp + 4, lp + 4);
  }
#endif
  __syncthreads();

  const int cn = tid >> 4;   // column handled in reduction phases
  const int cj = tid & 15;   // base row offset

  for (int t = 0; t < SEQLEN; ++t) {
    // ---- phase A: per-column max, exp shift, per-column sum ----
    float r[8];
    float mx = -3.0e38f;
#pragma unroll
    for (int q = 0; q < 8; ++q) {
      r[q] = fv[cn * COLSTR + cj + 16 * q];
      mx = fmaxf(mx, r[q]);
    }
    mx = hmax16(mx);
    float s = 0.0f;
#pragma unroll
    for (int q = 0; q < 8; ++q) {
      const float e = __expf(r[q] - mx);
      vb[cn * COLSTR + cj + 16 * q] = e;
      s += e;
    }
    s = hsum16(s);
    if (cj == 0) { Mcol[cn] = mx; Scol[cn] = s; }
    __syncthreads();

#if HAVE_ASYNC_LDS
    // kick off next step's emission tile while the WMMAs run
    if (t + 1 < SEQLEN) {
      const float* gp = inputs + ((size_t)(b0 + ln) * SEQLEN + (t + 1)) * TAGS + irow;
      float* lp = &em[((((size_t)(t + 1) & 1) * 8 + wave) * NB + ln) * ESTR + 8 * lh];
      ASYNC_B128(gp,     lp);
      ASYNC_B128(gp + 4, lp + 4);
    }
#endif

    // ---- phase B: Y = E_tile (16x128) x V (128x16) ----
    // Preload all B fragments (64 VGPRs) so WMMAs issue without DS stalls.
    v2f Bv[32];
#pragma unroll
    for (int kt = 0; kt < 32; ++kt) {
      // B-layout (4x16): lane n = ln, vgpr r holds K = 2*lh + r
      Bv[kt] = *(const v2f*)&vb[ln * COLSTR + 4 * kt + 2 * lh];
    }
    const v8f z = {0.f, 0.f, 0.f, 0.f, 0.f, 0.f, 0.f, 0.f};
    v8f acc[4] = {z, z, z, z};
#pragma unroll
    for (int kt = 0; kt < 32; ++kt) {
      acc[kt & 3] = __builtin_amdgcn_wmma_f32_16x16x4_f32(
          false, A[kt], false, Bv[kt], (short)0, acc[kt & 3], false, false);
    }
    const v8f y8 = (acc[0] + acc[1]) + (acc[2] + acc[3]);

    // ---- phase C: fv_new = emit + M + log(Y)  (mask==0 -> logsumexp(fv)) ----
    const int   mcol = (mbits[ln * (SEQLEN / 32) + (t >> 5)] >> (t & 31)) & 1;
    const float Mn   = Mcol[ln];
    const float lsS  = __logf(Scol[ln]) + Mn;

    float em8[8];
#if HAVE_ASYNC_LDS
    if (t + 1 < SEQLEN) { WAIT_ASYNC(2); } else { WAIT_ASYNC(0); }
    {
      const float* lp = &em[((((size_t)t & 1) * 8 + wave) * NB + ln) * ESTR + 8 * lh];
      const float4 e0 = *(const float4*)(lp);
      const float4 e1 = *(const float4*)(lp + 4);
      em8[0]=e0.x; em8[1]=e0.y; em8[2]=e0.z; em8[3]=e0.w;
      em8[4]=e1.x; em8[5]=e1.y; em8[6]=e1.z; em8[7]=e1.w;
    }
#else
    {
      const float* ep = inputs + ((size_t)(b0 + ln) * SEQLEN + t) * TAGS + irow;
      if (t + 1 < SEQLEN) __builtin_prefetch(ep + TAGS, 0, 0);
      const float4 e0 = *(const float4*)(ep);
      const float4 e1 = *(const float4*)(ep + 4);
      em8[0]=e0.x; em8[1]=e0.y; em8[2]=e0.z; em8[3]=e0.w;
      em8[4]=e1.x; em8[5]=e1.y; em8[6]=e1.z; em8[7]=e1.w;
    }
#endif

#pragma unroll
    for (int q = 0; q < 8; ++q) {
      // C/D layout: vgpr q = row (q + 8*lh), lane ln = column
      const float nv = (mcol != 0)
                         ? em8[q] + Mn + fmaxf(__logf(y8[q]), -1.0e30f)
                         : lsS;
      fv[ln * COLSTR + irow + q] = nv;
    }
    __syncthreads();
  }

  // ---- terminal: log_den[col] = logsumexp_i( fv[i] + trans[STOP, i] ) ----
  float r2[8];
  float mx = -3.0e38f;
#pragma unroll
  for (int q = 0; q < 8; ++q) {
    const int j = cj + 16 * q;
    r2[q] = fv[cn * COLSTR + j] + trans[STOP_TAG * TAGS + j];
    mx = fmaxf(mx, r2[q]);
  }
  mx = hmax16(mx);
  float s = 0.0f;
#pragma unroll
  for (int q = 0; q < 8; ++q) s += __expf(r2[q] - mx);
  s = hsum16(s);
  if (cj == 0) log_den[b0 + cn] = mx + __logf(s);
}

// ---------------------------------------------------------------------------
// Numerator: one wave per batch element, lanes strided over time.
// ---------------------------------------------------------------------------
__global__ __launch_bounds__(32)
void crf_numerator(const float* __restrict__ inputs,
                   const int*   __restrict__ tags,
                   const int*   __restrict__ mask,
                   const float* __restrict__ trans,
                   float*       __restrict__ log_num) {
  const int b    = blockIdx.x;
  const int lane = threadIdx.x;
  const int*   tg = tags + (size_t)b * SEQLEN;
  const int*   mk = mask + (size_t)b * SEQLEN;
  const float* in = inputs + (size_t)b * SEQLEN * TAGS;
  float acc = 0.0f;
  for (int t = lane; t < SEQLEN - 1; t += 32) {
    const int   ct = tg[t];
    const int   nt = tg[t + 1];
    const float mf = (float)mk[t];
    acc += (trans[nt * TAGS + ct] + in[(size_t)t * TAGS + ct]) * mf;
  }
#pragma unroll
  for (int m = 1; m < 32; m <<= 1) acc += __shfl_xor(acc, m, 32);
  if (lane == 0) {
    const int tL = tg[SEQLEN - 1];
    log_num[b] = acc
               + trans[tg[0] * TAGS + START_TAG]
               + trans[STOP_TAG * TAGS + tL]
               + in[(size_t)(SEQLEN - 1) * TAGS + tL] * (float)mk[SEQLEN - 1];
  }
}

// ---------------------------------------------------------------------------
// Finalize: out = sum_b (log_num[b] - log_den[b])
// ---------------------------------------------------------------------------
__global__ __launch_bounds__(256)
void crf_finalize(const float* __restrict__ log_num,
                  const float* __restrict__ log_den,
                  float*       __restrict__ out) {
  __shared__ float red[8];
  const int tid = threadIdx.x;
  float v = log_num[tid] - log_den[tid];
#pragma unroll
  for (int m = 1; m < 32; m <<= 1) v += __shfl_xor(v, m, 32);
  if ((tid & 31) == 0) red[tid >> 5] = v;
  __syncthreads();
  if (tid < 8) {
    float w = red[tid];
#pragma unroll
    for (int m = 1; m < 8; m <<= 1) w += __shfl_xor(w, m, 32);
    if (tid == 0) out[0] = w;
  }
}

extern "C" void kernel_launch(void* const* d_in, const int* in_sizes, int n_in,
                              void* d_out, int out_size, void* d_ws, size_t ws_size,
                              hipStream_t stream) {
  (void)in_sizes; (void)n_in; (void)out_size; (void)ws_size;
  const float* inputs = (const float*)d_in[0];
  const int*   tags   = (const int*)d_in[1];
  const int*   mask   = (const int*)d_in[2];
  const float* trans  = (const float*)d_in[3];

  float* num = (float*)d_ws;          // BATCH floats
  float* den = num + BATCH;           // BATCH floats

  crf_numerator   <<<BATCH, 32, 0, stream>>>(inputs, tags, mask, trans, num);
  crf_forward_scan<<<BATCH / NB, 256, 0, stream>>>(inputs, mask, trans, den);
  crf_finalize    <<<1, 256, 0, stream>>>(num, den, (float*)d_out);
}